// QLayer_28621662060686
// MI455X (gfx1250) — compile-verified
//
#include <hip/hip_runtime.h>
#include <hip/hip_bf16.h>

typedef __attribute__((ext_vector_type(2))) float v2f;
typedef __attribute__((ext_vector_type(4))) float v4f;
typedef __attribute__((ext_vector_type(8))) float v8f;

#define NW 24
#define PADR 20   // LDS row stride in floats (16 data + 4 pad, keeps 16B align)

// ---------------------------------------------------------------------------
// Build fused 16x16 stage matrices: M_b = C(4j+2) C(4j+1) C(4j) X_b R4 (kron)
// Layout: mats[((k*6+j)*2+b)*256 + p*16 + ip]  (row-major, output row p)
// Group bit convention: wire 4j -> bit3 of the 4-bit group index.
// ---------------------------------------------------------------------------
__global__ void build_mats(const float* __restrict__ angles,
                           float* __restrict__ mats) {
  int bid = blockIdx.x;            // 0..95
  int k = bid / 12;
  int j = (bid % 12) >> 1;
  int b = bid & 1;
  int e  = threadIdx.x;            // 0..255
  int i  = e >> 4;                 // pre-permutation row
  int ip = e & 15;                 // input column
  float r = 1.0f;
#pragma unroll
  for (int t = 0; t < 4; ++t) {
    float th = angles[k * NW + 4 * j + t];
    float c = cosf(0.5f * th);
    float s = sinf(0.5f * th);
    int bi = (i  >> (3 - t)) & 1;
    int bj = (ip >> (3 - t)) & 1;
    float v = bi ? (bj ? c : s) : (bj ? -s : c);
    r *= v;
  }
  int i3 = (i >> 3) & 1, i2 = (i >> 2) & 1, i1 = (i >> 1) & 1, i0 = i & 1;
  if (j > 0 && b) i3 ^= 1;         // boundary CNOT (conditioned X), then chain
  i2 ^= i3; i1 ^= i2; i0 ^= i1;
  int p = (i3 << 3) | (i2 << 2) | (i1 << 1) | i0;
  mats[(((k * 6 + j) * 2 + b) << 8) + (p << 4) + ip] = r;
}

__global__ void init_state(float* __restrict__ s, float* __restrict__ sum) {
  unsigned i = blockIdx.x * 256u + threadIdx.x;
  s[i] = (i == 0u) ? 1.0f : 0.0f;
  if (i == 0u) *sum = 0.0f;
}

// ---------------------------------------------------------------------------
// Fused pair kernel: applies stage j (bits gb+4..gb+7) then stage j+1
// (bits gb..gb+3) to a 256-row x 16-col LDS-resident tile.
// PAIR: 0 -> gb=16 (stages 0,1), 1 -> gb=8 (stages 2,3), 2 -> gb=0 (stages 4,5)
// ---------------------------------------------------------------------------
template <int PAIR>
__global__ void fused_pair(float* __restrict__ state,
                           const float* __restrict__ matsHi2,
                           const float* __restrict__ matsLo2) {
  __shared__ float L[256 * PADR];
  const int tid = threadIdx.x;
  const unsigned T = blockIdx.x;   // 4096 workgroups

  // ---- cooperative coalesced load into LDS: L[r8*PADR + t] ----
  unsigned bHi;
  unsigned gbase0, gbase1;         // reused for store
  if (PAIR < 2) {
    const int gb = (PAIR == 0) ? 16 : 8;
    unsigned cb = T << 4;          // column base, multiple of 16
    unsigned tileBase = ((cb >> gb) << (gb + 8)) | (cb & ((1u << gb) - 1u));
    bHi = (cb >> gb) & 1u;         // stage-j boundary bit (0 for stage 0)
    gbase0 = tileBase + ((unsigned)tid << gb);   // this thread's run (row r8=tid)
    const v4f* gp = (const v4f*)(state + gbase0);
    v4f x0 = gp[0], x1 = gp[1], x2 = gp[2], x3 = gp[3];
    v4f* Lr = (v4f*)(L + tid * PADR);
    Lr[0] = x0; Lr[1] = x1; Lr[2] = x2; Lr[3] = x3;
    gbase1 = 0;
  } else {
    unsigned par = T & 1u, cg = T >> 1;
    bHi = par;                     // stage-4 boundary bit
    unsigned t  = (unsigned)tid >> 4;          // column 0..15
    unsigned r0 = ((unsigned)tid & 15u) << 4;  // 16-row chunk
    gbase0 = cg * 8192u + par * 256u + t * 512u + r0;
    gbase1 = t;                    // remember column for store
    const v4f* gp = (const v4f*)(state + gbase0);
    v4f x0 = gp[0], x1 = gp[1], x2 = gp[2], x3 = gp[3];
#pragma unroll
    for (int e = 0; e < 4; ++e) {
      L[(r0 + e) * PADR + t]      = x0[e];
      L[(r0 + 4 + e) * PADR + t]  = x1[e];
      L[(r0 + 8 + e) * PADR + t]  = x2[e];
      L[(r0 + 12 + e) * PADR + t] = x3[e];
    }
  }
  __syncthreads();

  const int lane  = tid & 31;
  const int wave  = tid >> 5;
  const int n     = lane & 15;           // col within sub-tile; also A row m
  const int koff  = (lane >> 4) * 2;     // wave32 WMMA K-pair select
  const int mbase = (lane >> 4) * 8;     // D rows held by this lane

  // ---- phase A: 16x16 matrix over rHi; sub-tile = fixed rLo ----
  {
    const float* UH = matsHi2 + (bHi << 8);
#pragma unroll
    for (int si = 0; si < 2; ++si) {
      int sub = wave + 8 * si;           // rLo
      v8f acc = {};
#pragma unroll
      for (int q = 0; q < 4; ++q) {
        int kk = 4 * q + koff;           // rHi (K index)
        v2f a, bf;
        a[0]  = UH[n * 16 + kk];
        a[1]  = UH[n * 16 + kk + 1];
        bf[0] = L[((kk << 4) | sub) * PADR + n];
        bf[1] = L[(((kk + 1) << 4) | sub) * PADR + n];
        acc = __builtin_amdgcn_wmma_f32_16x16x4_f32(
            false, a, false, bf, (short)0, acc, false, false);
      }
#pragma unroll
      for (int g = 0; g < 8; ++g)
        L[(((mbase + g) << 4) | sub) * PADR + n] = acc[g];
    }
  }
  __syncthreads();

  // ---- phase B: 16x16 matrix over rLo; sub-tile = fixed rHi ----
  {
#pragma unroll
    for (int si = 0; si < 2; ++si) {
      int sub = wave + 8 * si;           // rHi
      const float* UL = matsLo2 + ((unsigned)(sub & 1) << 8);  // boundary = bit gb+4
      v8f acc = {};
#pragma unroll
      for (int q = 0; q < 4; ++q) {
        int kk = 4 * q + koff;           // rLo (K index)
        v2f a, bf;
        a[0]  = UL[n * 16 + kk];
        a[1]  = UL[n * 16 + kk + 1];
        bf[0] = L[((sub << 4) | kk) * PADR + n];
        bf[1] = L[((sub << 4) | (kk + 1)) * PADR + n];
        acc = __builtin_amdgcn_wmma_f32_16x16x4_f32(
            false, a, false, bf, (short)0, acc, false, false);
      }
#pragma unroll
      for (int g = 0; g < 8; ++g)
        L[((sub << 4) | (mbase + g)) * PADR + n] = acc[g];
    }
  }
  __syncthreads();

  // ---- cooperative coalesced store from LDS ----
  if (PAIR < 2) {
    const v4f* Lr = (const v4f*)(L + tid * PADR);
    v4f y0 = Lr[0], y1 = Lr[1], y2 = Lr[2], y3 = Lr[3];
    v4f* gp = (v4f*)(state + gbase0);
    gp[0] = y0; gp[1] = y1; gp[2] = y2; gp[3] = y3;
  } else {
    unsigned t  = gbase1;
    unsigned r0 = ((unsigned)tid & 15u) << 4;
    v4f y0, y1, y2, y3;
#pragma unroll
    for (int e = 0; e < 4; ++e) {
      y0[e] = L[(r0 + e) * PADR + t];
      y1[e] = L[(r0 + 4 + e) * PADR + t];
      y2[e] = L[(r0 + 8 + e) * PADR + t];
      y3[e] = L[(r0 + 12 + e) * PADR + t];
    }
    v4f* gp = (v4f*)(state + gbase0);
    gp[0] = y0; gp[1] = y1; gp[2] = y2; gp[3] = y3;
  }
}

// ---------------------------------------------------------------------------
// x = (0.6*tanh(0.1*2^23*amp^2))^0.15, accumulate sum; then subtract mean.
// ---------------------------------------------------------------------------
__global__ void finalize_x(float* __restrict__ x, float* __restrict__ sum) {
  __shared__ float red[256];
  unsigned i = blockIdx.x * 256u + threadIdx.x;
  float v = x[i];
  float t = tanhf(0.1f * 8388608.0f * (v * v));
  float y = powf(0.6f * t, 0.15f);
  x[i] = y;
  red[threadIdx.x] = y;
  __syncthreads();
  for (int s = 128; s > 0; s >>= 1) {
    if ((int)threadIdx.x < s) red[threadIdx.x] += red[threadIdx.x + s];
    __syncthreads();
  }
  if (threadIdx.x == 0) atomicAdd(sum, red[0]);
}

__global__ void finalize_sub(float* __restrict__ x,
                             const float* __restrict__ sum) {
  unsigned i = blockIdx.x * 256u + threadIdx.x;
  x[i] -= (*sum) * (1.0f / 16777216.0f);
}

extern "C" void kernel_launch(void* const* d_in, const int* in_sizes, int n_in,
                              void* d_out, int out_size, void* d_ws,
                              size_t ws_size, hipStream_t stream) {
  const float* angles = (const float*)d_in[0];     // 8 x 24 f32
  float* state = (float*)d_out;                    // 2^24 f32: state, then output
  float* mats  = (float*)d_ws;                     // 96 * 256 f32
  float* sum   = mats + 96 * 256;
  const unsigned NSTATE = 1u << 24;

  build_mats<<<96, 256, 0, stream>>>(angles, mats);
  init_state<<<NSTATE / 256, 256, 0, stream>>>(state, sum);

  for (int k = 0; k < 8; ++k) {
    const float* mk = mats + (k * 6) * 2 * 256;
    fused_pair<0><<<4096, 256, 0, stream>>>(state, mk + 0 * 512, mk + 1 * 512);
    fused_pair<1><<<4096, 256, 0, stream>>>(state, mk + 2 * 512, mk + 3 * 512);
    fused_pair<2><<<4096, 256, 0, stream>>>(state, mk + 4 * 512, mk + 5 * 512);
  }

  finalize_x<<<NSTATE / 256, 256, 0, stream>>>(state, sum);
  finalize_sub<<<NSTATE / 256, 256, 0, stream>>>(state, sum);
}